// URRobot_42915313221698
// MI455X (gfx1250) — compile-verified
//
#include <hip/hip_runtime.h>
#include <hip/hip_bf16.h>

// ---------------------------------------------------------------------------
// UR robot forward kinematics.
//   pose  [B,4,4] f32, theta [B,6] f32, verts/norms [7,4000,4] f32
//   out   [B,14,4000,3] f32   (7 vert channels then 7 norm channels)
//
// Phase 1: per-batch kinematic chain -> T_vis rows 0..2  (B*7*12 f32 in d_ws)
// Phase 2: bulk point transform via V_WMMA_F32_16X16X4_F32 (K=4 = homogeneous
//          coords). Fully unrolled 8-subtile fast path (8 A-loads in flight,
//          8 back-to-back WMMAs), LDS re-staging, coalesced NON-TEMPORAL b128
//          stores (344 MB write-once output must not evict the L2-resident
//          verts/norms/tvis working set; store-BW floor ~15 us at 23.3 TB/s).
// ---------------------------------------------------------------------------

#define PI_F 3.14159265358979323846f

// (roll, pitch, yaw, x, y, z) per link
__constant__ float KIN_P[7][6] = {
    {0.f, 0.f, PI_F, 0.f, 0.f, 0.f},
    {0.f, 0.f, 0.f, 0.f, 0.f, 0.15185f},
    {PI_F * 0.5f, 0.f, 0.f, 0.f, 0.f, 0.f},
    {0.f, 0.f, 0.f, -0.24355f, 0.f, 0.f},
    {0.f, 0.f, 0.f, -0.2132f, 0.f, 0.13105f},
    {PI_F * 0.5f, 0.f, 0.f, 0.f, -0.08535f, 0.f},
    {PI_F * 0.5f, PI_F, PI_F, 0.f, 0.0921f, 0.f}};

__constant__ float VIS_P[7][6] = {
    {0.f, 0.f, PI_F, 0.f, 0.f, 0.f},
    {0.f, 0.f, PI_F, 0.f, 0.f, 0.f},
    {PI_F * 0.5f, 0.f, -PI_F * 0.5f, 0.f, 0.f, 0.12f},
    {PI_F * 0.5f, 0.f, -PI_F * 0.5f, 0.f, 0.f, 0.027f},
    {PI_F * 0.5f, 0.f, 0.f, 0.f, 0.f, -0.104f},
    {0.f, 0.f, 0.f, 0.f, 0.f, -0.08535f},
    {PI_F * 0.5f, 0.f, 0.f, 0.f, 0.f, -0.0921f}};

// T = Tr(x,y,z) * Rz(yaw) * Ry(pitch) * Rx(roll), row-major 4x4
__device__ inline void t_origin(const float* p, float* T) {
    float sr, cr, sp, cp, sy, cy;
    sincosf(p[0], &sr, &cr);
    sincosf(p[1], &sp, &cp);
    sincosf(p[2], &sy, &cy);
    T[0] = cy * cp; T[1] = cy * sp * sr - sy * cr; T[2] = cy * sp * cr + sy * sr; T[3] = p[3];
    T[4] = sy * cp; T[5] = sy * sp * sr + cy * cr; T[6] = sy * sp * cr - cy * sr; T[7] = p[4];
    T[8] = -sp;     T[9] = cp * sr;                T[10] = cp * cr;               T[11] = p[5];
    T[12] = 0.f; T[13] = 0.f; T[14] = 0.f; T[15] = 1.f;
}

__device__ inline void mm4(const float* A, const float* B, float* C) {
#pragma unroll
    for (int i = 0; i < 4; ++i)
#pragma unroll
        for (int j = 0; j < 4; ++j)
            C[i * 4 + j] = fmaf(A[i * 4 + 0], B[0 * 4 + j],
                           fmaf(A[i * 4 + 1], B[1 * 4 + j],
                           fmaf(A[i * 4 + 2], B[2 * 4 + j],
                                A[i * 4 + 3] * B[3 * 4 + j])));
}

// ------------------------- Phase 1: chain + T_vis --------------------------
__global__ void fk_transforms(const float* __restrict__ pose,
                              const float* __restrict__ theta,
                              float* __restrict__ tvis, int B) {
    int b = blockIdx.x * blockDim.x + threadIdx.x;
    if (b >= B) return;

    float t[16], tmp[16], K[16], R[16], M[16];

    // t = pose[b] @ KIN_T[0]
    t_origin(KIN_P[0], K);
    mm4(pose + b * 16, K, t);

#pragma unroll
    for (int l = 0; l < 7; ++l) {
        if (l > 0) {
            // rz(theta[l-1])
            float s, c;
            sincosf(theta[b * 6 + (l - 1)], &s, &c);
#pragma unroll
            for (int q = 0; q < 16; ++q) R[q] = 0.f;
            R[0] = c; R[1] = -s; R[4] = s; R[5] = c; R[10] = 1.f; R[15] = 1.f;
            t_origin(KIN_P[l], K);
            mm4(K, R, tmp);          // tt = KIN_T[l] @ Rz
            mm4(t, tmp, M);          // chain
#pragma unroll
            for (int q = 0; q < 16; ++q) t[q] = M[q];
        }
        // T_vis rows 0..2 (row 3 never contributes to output[..., :3])
        t_origin(VIS_P[l], K);
        mm4(t, K, M);
        float* dst = tvis + (b * 7 + l) * 12;
#pragma unroll
        for (int q = 0; q < 12; ++q) dst[q] = M[q];
    }
}

// ------------------------- Phase 2: bulk WMMA ------------------------------
typedef float v2f __attribute__((ext_vector_type(2)));
typedef float v4f __attribute__((ext_vector_type(4)));
typedef float v8f __attribute__((ext_vector_type(8)));

#define PTS 128  // points per wave-tile (8 WMMA sub-tiles of 16)

// Grid: x = channel (0..13), y = point-tile, z = batch-group-octet.
// Block = 256 threads = 8 waves; wave w handles batch group (z*8+w), 4 batches.
// A (16x4 f32): lane m=L&15, VGPR0 holds K=(half?2:0), VGPR1 K=(half?3:1).
// B (4x16 f32): column n packs (batch_local = n/3, comp i = n%3); same K split.
// D (16x16 f32): VGPR r -> M = r + 8*half, N = lane&15.
__global__ void __launch_bounds__(256) fk_points(
    const float* __restrict__ verts, const float* __restrict__ norms,
    const float* __restrict__ tvis, float* __restrict__ out, int B, int V) {
    __shared__ float stage[8][4 * PTS * 3];  // 48 KB: per-wave staging

    const int lane = threadIdx.x & 31;
    const int wave = threadIdx.x >> 5;   // wave32
    const int c = blockIdx.x;            // 0..13
    const int l = (c < 7) ? c : (c - 7);
    const float* __restrict__ src = (c < 7) ? verts : norms;
    const int v0 = blockIdx.y * PTS;
    const int b_base = (blockIdx.z * 8 + wave) * 4;

    const int n = lane & 15;
    const int kh = lane >> 4;

    // --- B operand: rows of T_vis for 4 batches, cols 12..15 = 0 ---
    v2f bmat; bmat.x = 0.f; bmat.y = 0.f;
    const bool bvalid = (n < 12) && (b_base + n / 3 < B);
    if (bvalid) {
        const float* row = tvis + (((b_base + n / 3) * 7 + l) * 3 + (n % 3)) * 4;
        bmat.x = row[2 * kh];
        bmat.y = row[2 * kh + 1];
    }

    int nsub = (V - v0) / 16;  // V % 16 == 0 (4000)
    if (nsub > 8) nsub = 8;
    float* st = stage[wave];
    const float* abase = src + (size_t)((l * V + v0 + n) * 4 + 2 * kh);

    if (nsub == 8) {
        // ---- fast path: all 8 A-loads issued before any WMMA consumes ----
        v2f a[8];
#pragma unroll
        for (int s = 0; s < 8; ++s) {
            const float* arow = abase + s * 64;  // 16 rows * 4 floats
            a[s].x = arow[0];
            a[s].y = arow[1];
        }
        v8f d[8];
#pragma unroll
        for (int s = 0; s < 8; ++s) {
            v8f z = {};
            d[s] = __builtin_amdgcn_wmma_f32_16x16x4_f32(
                false, a[s], false, bmat, (short)0, z, false, false);
        }
        if (n < 12) {
            const int j = n / 3, i = n % 3;
            float* sb = st + (j * PTS) * 3 + i;
#pragma unroll
            for (int s = 0; s < 8; ++s)
#pragma unroll
                for (int r = 0; r < 8; ++r)
                    sb[(s * 16 + r + 8 * kh) * 3] = d[s][r];
        }
    } else {
        // ---- ragged tail tile ----
        for (int s = 0; s < nsub; ++s) {
            const float* arow = abase + s * 64;
            v2f amat; amat.x = arow[0]; amat.y = arow[1];
            v8f z = {};
            v8f dd = __builtin_amdgcn_wmma_f32_16x16x4_f32(
                false, amat, false, bmat, (short)0, z, false, false);
            if (n < 12) {
                const int j = n / 3, i = n % 3;
#pragma unroll
                for (int r = 0; r < 8; ++r)
                    st[(j * PTS + s * 16 + r + 8 * kh) * 3 + i] = dd[r];
            }
        }
    }
    __builtin_amdgcn_wave_barrier();  // same-wave DS ordering fence for compiler

    // --- coalesced non-temporal b128 stores: 512 B contiguous per store ---
    if (nsub == 8) {
#pragma unroll
        for (int j = 0; j < 4; ++j) {
            if (b_base + j >= B) break;
            const long long base = (((long long)(b_base + j) * 14 + c) * V + v0) * 3;
            v4f* __restrict__ dst = (v4f*)(out + base);
            const v4f* s4 = (const v4f*)(st + j * PTS * 3);
#pragma unroll
            for (int it = 0; it < 3; ++it) {
                const int idx = it * 32 + lane;
                __builtin_nontemporal_store(s4[idx], &dst[idx]);
            }
        }
    } else {
        const int nfl4 = nsub * 12;
#pragma unroll
        for (int j = 0; j < 4; ++j) {
            if (b_base + j >= B) break;
            const long long base = (((long long)(b_base + j) * 14 + c) * V + v0) * 3;
            v4f* __restrict__ dst = (v4f*)(out + base);
            const v4f* s4 = (const v4f*)(st + j * PTS * 3);
            for (int it = lane; it < nfl4; it += 32)
                __builtin_nontemporal_store(s4[it], &dst[it]);
        }
    }
}

extern "C" void kernel_launch(void* const* d_in, const int* in_sizes, int n_in,
                              void* d_out, int out_size, void* d_ws, size_t ws_size,
                              hipStream_t stream) {
    const float* pose  = (const float*)d_in[0];  // [B,4,4]
    const float* theta = (const float*)d_in[1];  // [B,6]
    const float* verts = (const float*)d_in[2];  // [7,V,4]
    const float* norms = (const float*)d_in[3];  // [7,V,4]
    float* out = (float*)d_out;
    float* tvis = (float*)d_ws;                  // [B,7,3,4] f32

    const int B = in_sizes[0] / 16;              // 512
    const int V = in_sizes[2] / (7 * 4);         // 4000

    fk_transforms<<<(B + 255) / 256, 256, 0, stream>>>(pose, theta, tvis, B);

    dim3 grid(14, (V + PTS - 1) / PTS, (B + 31) / 32);
    fk_points<<<grid, 256, 0, stream>>>(verts, norms, tvis, out, B, V);
}